// NGCF_91182155694432
// MI455X (gfx1250) — compile-verified
//
#include <hip/hip_runtime.h>
#include <hip/hip_bf16.h>

typedef __attribute__((ext_vector_type(2))) float v2f;
typedef __attribute__((ext_vector_type(8))) float v8f;

#define N_USER 50000
#define N_ITEM 100000
#define N_NODES 150000
#define NPAD 150016L   // padded to multiple of 32 rows for WMMA tiling
#define DD 64
#define NNZ 2400000

// ---------------------------------------------------------------------------
// Build E = concat(user_emb, item_emb), zero padded rows. float4 copies.
// ---------------------------------------------------------------------------
__global__ __launch_bounds__(256) void ngcf_init_E(const float* __restrict__ ue,
                                                   const float* __restrict__ ie,
                                                   float* __restrict__ E) {
    long idx = (long)blockIdx.x * blockDim.x + threadIdx.x;  // float4 index
    long e = idx * 4;
    if (e >= NPAD * DD) return;
    long row = e >> 6;
    float4 v;
    if (row < N_USER)       v = *(const float4*)(ue + e);
    else if (row < N_NODES) v = *(const float4*)(ie + (e - (long)N_USER * DD));
    else                    v = make_float4(0.f, 0.f, 0.f, 0.f);
    *(float4*)(E + e) = v;
}

// ---------------------------------------------------------------------------
// user_mlp = (feat@lin1_w + lin1_b)@lin2_w + lin2_b ; scatter-blend into E.
// One block (64 threads) per batch row.
// ---------------------------------------------------------------------------
__global__ __launch_bounds__(64) void ngcf_mlp_update(
    const float* __restrict__ user_emb, const float* __restrict__ user_feat,
    const float* __restrict__ l1w, const float* __restrict__ l1b,
    const float* __restrict__ l2w, const float* __restrict__ l2b,
    const int* __restrict__ user_idx, const float* __restrict__ ratio_p,
    float* __restrict__ E) {
    __shared__ float hid[32];
    const int b = blockIdx.x;
    const int t = threadIdx.x;
    if (t < 32) {
        float h = l1b[t];
#pragma unroll
        for (int f = 0; f < 4; ++f) h += user_feat[b * 4 + f] * l1w[f * 32 + t];
        hid[t] = h;
    }
    __syncthreads();
    float mval = l2b[t];
#pragma unroll
    for (int hI = 0; hI < 32; ++hI) mval += hid[hI] * l2w[hI * DD + t];
    const float ratio = ratio_p[0];
    const long u = user_idx[b];
    E[u * DD + t] = user_emb[u * DD + t] * (1.0f - ratio) + mval * ratio;
}

// ---------------------------------------------------------------------------
// SpMM scatter: LE[row] += val * E[col].  One wave per nnz (grid-stride),
// lane covers 2 of 64 columns via float2; hardware f32 atomics into L2-resident
// LE.  Prefetch next gathered row.
// ---------------------------------------------------------------------------
__global__ __launch_bounds__(256) void ngcf_spmm(
    const float* __restrict__ E, const int* __restrict__ rows,
    const int* __restrict__ cols, const float* __restrict__ vals,
    float* __restrict__ LE, int nnz) {
    const int lane = threadIdx.x & 31;
    int w = blockIdx.x * (blockDim.x >> 5) + (threadIdx.x >> 5);
    const int nw = gridDim.x * (blockDim.x >> 5);
    for (int e = w; e < nnz; e += nw) {
        const int r = rows[e];
        const int c = cols[e];
        const float v = vals[e];
        if (e + nw < nnz) {  // prefetch next gather target row into caches
            __builtin_prefetch(E + (long)cols[e + nw] * DD, 0, 1);
        }
        const float2 ev = *(const float2*)(E + (long)c * DD + lane * 2);
        float* dst = LE + (long)r * DD + lane * 2;
        unsafeAtomicAdd(dst, v * ev.x);
        unsafeAtomicAdd(dst + 1, v * ev.y);
    }
}

// ---------------------------------------------------------------------------
// Fused layer GEMM via V_WMMA_F32_16X16X4_F32:
//   Eout = leaky_relu( (LE+E)@W1 + b1 + (LE*E)@W2 + b2 , 0.2 )
// Block = 256 threads = 8 waves; block owns 32 rows x 64 cols; each wave owns
// one 16x16 tile.  K=64 swept in 16 steps of 4; both GEMMs accumulate into one
// v8f accumulator (32 WMMAs / tile).  Weights staged in LDS.
// ---------------------------------------------------------------------------
__global__ __launch_bounds__(256) void ngcf_gemm_leaky(
    const float* __restrict__ E, const float* __restrict__ LE,
    const float* __restrict__ W1, const float* __restrict__ B1,
    const float* __restrict__ W2, const float* __restrict__ B2,
    float* __restrict__ Eout) {
    __shared__ float sw1[DD * DD];
    __shared__ float sw2[DD * DD];
    for (int i = threadIdx.x; i < DD * DD; i += 256) {
        sw1[i] = W1[i];
        sw2[i] = W2[i];
    }
    __syncthreads();

    const int wid = threadIdx.x >> 5;
    const int lane = threadIdx.x & 31;
    const int rowTile = wid >> 2;   // 0..1
    const int colTile = wid & 3;    // 0..3
    const long row0 = (long)blockIdx.x * 32 + rowTile * 16;
    const int col0 = colTile * 16;
    const int m = lane & 15;        // == n for B/C operands
    const int hi = lane >> 4;
    const int kb = hi * 2;

    const float* eRow = E + (row0 + m) * DD;
    const float* lRow = LE + (row0 + m) * DD;

    v8f acc = {};
#pragma unroll
    for (int kk = 0; kk < DD; kk += 4) {
        float2 ev = *(const float2*)(eRow + kk + kb);
        float2 lv = *(const float2*)(lRow + kk + kb);
        v2f a1, a2, bA, bB;
        a1.x = lv.x + ev.x;  a1.y = lv.y + ev.y;   // (L+I)E
        a2.x = lv.x * ev.x;  a2.y = lv.y * ev.y;   // LE (.) E
        bA.x = sw1[(kk + kb) * DD + col0 + m];
        bA.y = sw1[(kk + kb + 1) * DD + col0 + m];
        bB.x = sw2[(kk + kb) * DD + col0 + m];
        bB.y = sw2[(kk + kb + 1) * DD + col0 + m];
        acc = __builtin_amdgcn_wmma_f32_16x16x4_f32(false, a1, false, bA,
                                                    (short)0, acc, false, false);
        acc = __builtin_amdgcn_wmma_f32_16x16x4_f32(false, a2, false, bB,
                                                    (short)0, acc, false, false);
    }

    const float bias = B1[col0 + m] + B2[col0 + m];
#pragma unroll
    for (int j = 0; j < 8; ++j) {
        float x = acc[j] + bias;
        x = (x > 0.f) ? x : 0.2f * x;
        Eout[(row0 + j + 8 * hi) * DD + col0 + m] = x;
    }
}

// ---------------------------------------------------------------------------
// Gather 3x1024 requested rows into out slice [.,., layer*64 : layer*64+64],
// with optional fused L2 row-normalization (wave32 shfl reduction).
// One wave per output row; 384 blocks x 8 waves = 3072 rows.
// ---------------------------------------------------------------------------
__global__ __launch_bounds__(256) void ngcf_gather(
    const float* __restrict__ Ebuf, const int* __restrict__ user_idx,
    const int* __restrict__ pos_idx, const int* __restrict__ neg_idx,
    float* __restrict__ out, int layer, int do_norm) {
    const int lane = threadIdx.x & 31;
    const int w = blockIdx.x * 8 + (threadIdx.x >> 5);  // 0..3071
    const int s = w >> 10;
    const int r = w & 1023;
    long node;
    if (s == 0)      node = user_idx[r];
    else if (s == 1) node = N_USER + (long)pos_idx[r];
    else             node = N_USER + (long)neg_idx[r];
    float2 v = *(const float2*)(Ebuf + node * DD + lane * 2);
    if (do_norm) {
        float ss = v.x * v.x + v.y * v.y;
#pragma unroll
        for (int o = 16; o > 0; o >>= 1) ss += __shfl_xor(ss, o, 32);
        const float sc = 1.0f / fmaxf(sqrtf(ss), 1e-12f);
        v.x *= sc;
        v.y *= sc;
    }
    *(float2*)(out + (long)w * 256 + layer * DD + lane * 2) = v;
}

// ---------------------------------------------------------------------------
extern "C" void kernel_launch(void* const* d_in, const int* in_sizes, int n_in,
                              void* d_out, int out_size, void* d_ws, size_t ws_size,
                              hipStream_t stream) {
    const float* user_emb = (const float*)d_in[0];
    const float* item_emb = (const float*)d_in[1];
    const float* lin1_w   = (const float*)d_in[2];
    const float* lin1_b   = (const float*)d_in[3];
    const float* lin2_w   = (const float*)d_in[4];
    const float* lin2_b   = (const float*)d_in[5];
    const float* w1       = (const float*)d_in[6];   // [3][64][64]
    const float* b1       = (const float*)d_in[7];   // [3][64]
    const float* w2       = (const float*)d_in[8];
    const float* b2       = (const float*)d_in[9];
    const int*   lap_row  = (const int*)d_in[10];
    const int*   lap_col  = (const int*)d_in[11];
    const float* lap_val  = (const float*)d_in[12];
    const int*   user_idx = (const int*)d_in[13];
    const float* user_feat= (const float*)d_in[14];
    const int*   pos_idx  = (const int*)d_in[15];
    const int*   neg_idx  = (const int*)d_in[16];
    const float* ratio    = (const float*)d_in[17];

    const long bufElems = NPAD * DD;
    float* buf0 = (float*)d_ws;
    float* buf1 = buf0 + bufElems;
    float* bufL = buf1 + bufElems;

    float* out = (float*)d_out;

    // E0 = concat(user_emb with MLP-blended rows, item_emb)
    ngcf_init_E<<<(int)((NPAD * 16 + 255) / 256), 256, 0, stream>>>(user_emb, item_emb, buf0);
    ngcf_mlp_update<<<1024, 64, 0, stream>>>(user_emb, user_feat, lin1_w, lin1_b,
                                             lin2_w, lin2_b, user_idx, ratio, buf0);
    // layer-0 slice of the output (no normalization)
    ngcf_gather<<<384, 256, 0, stream>>>(buf0, user_idx, pos_idx, neg_idx, out, 0, 0);

    float* Ecur = buf0;
    float* Enext = buf1;
    for (int k = 0; k < 3; ++k) {
        hipMemsetAsync(bufL, 0, bufElems * sizeof(float), stream);
        ngcf_spmm<<<4096, 256, 0, stream>>>(Ecur, lap_row, lap_col, lap_val, bufL, NNZ);
        ngcf_gemm_leaky<<<(int)(NPAD / 32), 256, 0, stream>>>(
            Ecur, bufL, w1 + (long)k * DD * DD, b1 + (long)k * DD,
            w2 + (long)k * DD * DD, b2 + (long)k * DD, Enext);
        ngcf_gather<<<384, 256, 0, stream>>>(Enext, user_idx, pos_idx, neg_idx, out, k + 1, 1);
        float* t = Ecur; Ecur = Enext; Enext = t;
    }
}